// MLPRule_54674933678370
// MI455X (gfx1250) — compile-verified
//
#include <hip/hip_runtime.h>

typedef __attribute__((ext_vector_type(2))) float v2f;
typedef __attribute__((ext_vector_type(8))) float v8f;

#define N_IN   4096
#define N_OUT  4096
#define HIDDEN 10

// dW[i,j] = tanh( W2 . leaky_relu( W1 . [x_j, y_i, w_ij, r] + b1 ) + b2 )
//
// Layer 1 mapped to V_WMMA_F32_16X16X4_F32 as  h^T = W1(16x4 padded) * inp^T(4x16):
//   A (16x4 f32):  lane m(0..15): VGPR0=W1[m,0], VGPR1=W1[m,1];
//                  lane m+16:     VGPR0=W1[m,2], VGPR1=W1[m,3]   (rows 10..15 = 0)
//   B (4x16 f32):  VGPR0: lanes0-15 = row K=0 (x_j),  lanes16-31 = row K=2 (w_ij)
//                  VGPR1: lanes0-15 = row K=1 (y_i),  lanes16-31 = row K=3 (r)
//   C = b1 broadcast (rows >= 10 zero).
//   D: lane n holds synapse n; VGPR r holds hidden (r + 8*laneHalf).
// Layer 2 is 8 in-lane FMAs over the accumulator VGPRs + one xor-16 shuffle.
// B loads are software-pipelined one iteration ahead so the loadcnt wait sits
// behind a full WMMA+epilogue of independent work.
__global__ __launch_bounds__(256) void mlp_rule_wmma_kernel(
    const float* __restrict__ x,  const float* __restrict__ y,
    const float* __restrict__ w,  const float* __restrict__ r,
    const float* __restrict__ W1, const float* __restrict__ b1,
    const float* __restrict__ W2, const float* __restrict__ b2,
    float* __restrict__ out)
{
    const int row  = blockIdx.x;
    const int tid  = threadIdx.x;
    const int lane = tid & 31;
    const int wave = tid >> 5;
    const int l15  = lane & 15;
    const int half = lane >> 4;          // 0: lanes 0-15, 1: lanes 16-31

    // ---- loop-invariant operands -------------------------------------------
    v2f A;
    {
        const int  m     = l15;
        const int  f     = half * 2;
        const bool valid = (m < HIDDEN);
        A.x = valid ? W1[m * 4 + f]     : 0.0f;
        A.y = valid ? W1[m * 4 + f + 1] : 0.0f;
    }

    v8f   Cb;          // bias broadcast along N
    float w2c[8];      // layer-2 coefficients for this lane's hidden rows
    #pragma unroll
    for (int rr = 0; rr < 8; ++rr) {
        const int m = rr + half * 8;
        Cb[rr]  = (m < HIDDEN) ? b1[m] : 0.0f;
        w2c[rr] = (m < HIDDEN) ? W2[m] : 0.0f;
    }

    const float rv  = r[0];
    const float b2v = b2[0];
    const float yv  = y[row];
    const float* __restrict__ wrow = w   + (size_t)row * N_IN;
    float*       __restrict__ orow = out + (size_t)row * N_IN;

    // Per-lane constant row of B (K=1 -> y_i for lanes 0-15, K=3 -> r for 16-31)
    const float byv = half ? rv : yv;
    // Per-lane base for B's loaded row (K=0 -> x, K=2 -> this row of w)
    const float* __restrict__ bxbase = half ? wrow : x;

    const int ITERS   = (N_IN / 8) / 32;     // 16 iterations of 32 columns
    const int colBase = wave * (N_IN / 8);   // 512 columns per wave

    // ---- software pipeline: preload first iteration's B rows ---------------
    float bx0 = bxbase[colBase + l15];
    float bx1 = bxbase[colBase + 16 + l15];

    int j0 = colBase;
    for (int it = 0; it < ITERS; ++it, j0 += 32) {
        // Issue next iteration's loads now; waits land after this iteration's
        // WMMAs + epilogue. Last iteration re-reads colBase (clamped, no branch).
        const int jn = (it + 1 < ITERS) ? (j0 + 32) : colBase;
        const float nbx0 = bxbase[jn + l15];
        const float nbx1 = bxbase[jn + 16 + l15];

        // Keep the streamed w row flowing into L2 (global_prefetch_b8).
        __builtin_prefetch(bxbase + j0 + 256, 0, 1);

        float res[2];
        #pragma unroll
        for (int t = 0; t < 2; ++t) {
            v2f B;
            B.x = t ? bx1 : bx0;      // x_j (lanes 0-15) / w_ij (lanes 16-31)
            B.y = byv;                // y_i              / r

            // D = A*B + C : h^T tile, 16 synapses x 16 hidden(rows)
            v8f D = __builtin_amdgcn_wmma_f32_16x16x4_f32(
                        false, A, false, B, (short)0, Cb, false, false);

            // Layer 2: s = sum_r leaky(h_r) * W2_r  (in-lane over 8 VGPRs)
            float s = 0.0f;
            #pragma unroll
            for (int rr = 0; rr < 8; ++rr) {
                float h  = D[rr];
                float hl = fmaxf(h, 0.01f * h);        // leaky_relu, slope .01
                s = fmaf(hl, w2c[rr], s);
            }
            // combine hidden 8..9 held by the opposite half-wave
            s += __shfl_xor(s, 16, 32);

            // tanh(z) = (1 - e^(-2z)) / (1 + e^(-2z)), via v_exp_f32/v_rcp_f32
            float z = s + b2v;
            float e = __builtin_amdgcn_exp2f(z * -2.8853900817779268f); // -2*log2e
            res[t]  = 1.0f - 2.0f * e * __builtin_amdgcn_rcpf(1.0f + e);
        }

        // lanes 0-15 -> tile0 columns j0+lane, lanes 16-31 -> tile1 columns
        orow[j0 + lane] = half ? res[1] : res[0];

        bx0 = nbx0;   // s_wait_loadcnt lands here, behind a full iteration
        bx1 = nbx1;
    }
}

extern "C" void kernel_launch(void* const* d_in, const int* in_sizes, int n_in,
                              void* d_out, int out_size, void* d_ws, size_t ws_size,
                              hipStream_t stream) {
    (void)in_sizes; (void)n_in; (void)out_size; (void)d_ws; (void)ws_size;
    const float* x  = (const float*)d_in[0];
    const float* y  = (const float*)d_in[1];
    const float* w  = (const float*)d_in[2];
    const float* r  = (const float*)d_in[3];
    const float* W1 = (const float*)d_in[4];
    const float* b1 = (const float*)d_in[5];
    const float* W2 = (const float*)d_in[6];
    const float* b2 = (const float*)d_in[7];
    float* out = (float*)d_out;

    dim3 grid(N_OUT);   // one row per block
    dim3 block(256);    // 8 waves x 512 columns each
    mlp_rule_wmma_kernel<<<grid, block, 0, stream>>>(x, y, w, r, W1, b1, W2, b2, out);
}